// OctantSample_17042430231231
// MI455X (gfx1250) — compile-verified
//
#include <hip/hip_runtime.h>
#include <hip/hip_bf16.h>

// ---------------------------------------------------------------------------
// Octant counting-sort for [64, 3, 65536] point clouds -> [64, 8, 65536] i64.
// Bandwidth-bound (~312 MiB traffic -> ~13.4us at 23.3 TB/s). CDNA5 paths:
//   * Tensor Data Mover (tensor_load_to_lds + s_wait_tensorcnt) stages both
//     the strided [3 x CHUNK] f32 tile and the 1-D code tile into LDS.
//   * wave32 ballots (__builtin_amdgcn_ballot_w32) give exact, deterministic
//     descending ranks (no atomics in the position computation).
//   * Tail-only zero fill: stores total exactly 256 MiB (no double writes).
// ---------------------------------------------------------------------------

#define CHUNK 2048
#define BLOCK 256
#define PTS_PER_THREAD (CHUNK / BLOCK)   // 8
#define WAVES_PER_BLOCK (BLOCK / 32)     // 8
#define SEG 256                          // points per wave segment

typedef unsigned int v4u __attribute__((ext_vector_type(4)));
typedef int          v8i __attribute__((ext_vector_type(8)));
typedef int          v4i __attribute__((ext_vector_type(4)));

// TDM builtin arity differs between ROCm 7.2 (5 args) and the
// amdgpu-toolchain/therock headers (6 args). Hedge via header presence.
#if __has_include(<hip/amd_detail/amd_gfx1250_TDM.h>)
__device__ __forceinline__ void tdm_issue(v4u g0, v8i g1, v4i g2, v4i g3) {
    v8i g4 = {0, 0, 0, 0, 0, 0, 0, 0};
    __builtin_amdgcn_tensor_load_to_lds(g0, g1, g2, g3, g4, 0);
}
#else
__device__ __forceinline__ void tdm_issue(v4u g0, v8i g1, v4i g2, v4i g3) {
    __builtin_amdgcn_tensor_load_to_lds(g0, g1, g2, g3, 0);
}
#endif

// Build a D# for a (tile_d0 x tile_d1) tile of a row-major 2-D tensor and
// issue the DMA. data_size_code: 0=1B, 1=2B, 2=4B, 3=8B. tile_d1==0 => 1-D.
__device__ __forceinline__ void tdm_load_tile(const void* gptr, unsigned int lds_off,
                                              unsigned int data_size_code,
                                              unsigned int tensor_d0, unsigned int tensor_d1,
                                              unsigned int tile_d0, unsigned int tile_d1,
                                              unsigned int d0_stride) {
    unsigned long long gaddr = (unsigned long long)(uintptr_t)gptr;
    // Group 0: count=1 | lds_addr | global_addr[56:0] | type=2
    v4u g0 = { 1u,
               lds_off,
               (unsigned int)gaddr,
               (unsigned int)((gaddr >> 32) & 0x1FFFFFFu) | (2u << 30) };
    // Group 1: data_size | tensor_dim0 | tensor_dim1 | tile_dim0/1 | dim0_stride
    v8i g1 = { (int)(data_size_code << 16),
               (int)((tensor_d0 & 0xFFFFu) << 16),
               (int)((tensor_d0 >> 16) | ((tensor_d1 & 0xFFFFu) << 16)),
               (int)((tensor_d1 >> 16) | (tile_d0 << 16)),
               (int)tile_d1,                  // tile_dim1 lo16, tile_dim2=0
               (int)d0_stride,                // dim0_stride[31:0]
               0, 0 };
    v4i gz = { 0, 0, 0, 0 };
    tdm_issue(g0, g1, gz, gz);
}

// ---------------------------------------------------------------------------
// Kernel 1: TDM-stage [3 x CHUNK] f32 tile to LDS, emit octant codes +
// per-chunk 8-bin histogram.  grid = (nChunks, B), block = 256.
// ---------------------------------------------------------------------------
__global__ void octant_code_hist_kernel(const float* __restrict__ pcs,
                                        unsigned char* __restrict__ codes,
                                        unsigned int* __restrict__ hist,
                                        int N, int nChunks) {
    __shared__ __align__(16) float tile[3 * CHUNK];   // x-plane, y-plane, z-plane
    __shared__ unsigned int lhist[8];

    const int c = blockIdx.x;
    const int b = blockIdx.y;
    const int t = threadIdx.x;

    if (t < 8) lhist[t] = 0u;

    if (t < 32) {  // one wave issues the TDM DMA and waits on TENSORcnt
        const float* src = pcs + ((size_t)b * 3 * N + (size_t)c * CHUNK);
        tdm_load_tile(src, (unsigned int)(uintptr_t)&tile[0],
                      /*data_size=*/2u,
                      /*tensor_d0=*/(unsigned)N, /*tensor_d1=*/3u,
                      /*tile_d0=*/CHUNK, /*tile_d1=*/3u,
                      /*d0_stride=*/(unsigned)N);
        __builtin_amdgcn_s_wait_tensorcnt(0);
    }
    __syncthreads();

    const size_t codeBase = (size_t)b * N + (size_t)c * CHUNK;
#pragma unroll
    for (int k = 0; k < PTS_PER_THREAD; ++k) {
        const int il = t + k * BLOCK;
        const float x = tile[il];
        const float y = tile[CHUNK + il];
        const float z = tile[2 * CHUNK + il];
        const int o = ((x >= 0.0f) ? 4 : 0) | ((y >= 0.0f) ? 2 : 0) | ((z >= 0.0f) ? 1 : 0);
        codes[codeBase + il] = (unsigned char)o;
        atomicAdd(&lhist[o], 1u);
    }
    __syncthreads();

    if (t < 8) hist[((size_t)b * nChunks + c) * 8 + t] = lhist[t];
}

// ---------------------------------------------------------------------------
// Kernel 2: reverse exclusive scan over chunks per (b, o) + row totals.
// grid = (B), block = 32; threads 0..7 handle one octant each.
// ---------------------------------------------------------------------------
__global__ void rev_scan_kernel(const unsigned int* __restrict__ hist,
                                unsigned int* __restrict__ offs,
                                unsigned int* __restrict__ totals,
                                int nChunks) {
    const int b = blockIdx.x;
    const int t = threadIdx.x;
    if (t < 8) {
        unsigned int run = 0u;
        for (int c = nChunks - 1; c >= 0; --c) {
            offs[((size_t)b * 8 + t) * nChunks + c] = run;
            run += hist[((size_t)b * nChunks + c) * 8 + t];
        }
        totals[b * 8 + t] = run;   // members of octant t in batch b (incl. point 0)
    }
}

// ---------------------------------------------------------------------------
// Kernel 3: zero only the tail [cnt, N) of each octant row.
// grid = (N/CHUNK, 8, B), block = 256. Disjoint from the scatter writes.
// ---------------------------------------------------------------------------
__global__ void tail_fill_kernel(const unsigned int* __restrict__ totals,
                                 long long* __restrict__ out, int N) {
    const int s = blockIdx.x;           // slice within row
    const int o = blockIdx.y;
    const int b = blockIdx.z;
    const unsigned int cnt = totals[b * 8 + o];
    const size_t rowBase = ((size_t)b * 8 + o) * (size_t)N;
    const int start = s * CHUNK;
#pragma unroll
    for (int k = 0; k < PTS_PER_THREAD; ++k) {
        const unsigned int i = (unsigned int)(start + k * BLOCK + threadIdx.x);
        if (i >= cnt) out[rowBase + i] = 0ll;
    }
}

// ---------------------------------------------------------------------------
// Kernel 4: scatter each index to its exact descending-rank slot.
// grid = (nChunks, B), block = 256 (8 waves x 32 lanes, 256 pts per wave).
// ---------------------------------------------------------------------------
__global__ void scatter_ranks_kernel(const unsigned char* __restrict__ codes,
                                     const unsigned int* __restrict__ offs,
                                     long long* __restrict__ out,
                                     int N, int nChunks) {
    __shared__ __align__(16) unsigned char sc[CHUNK];
    __shared__ unsigned int segCnt[WAVES_PER_BLOCK][8];
    __shared__ unsigned int segBase[WAVES_PER_BLOCK][8];

    const int c = blockIdx.x;
    const int b = blockIdx.y;
    const int t = threadIdx.x;
    const int lane = t & 31;
    const int w = t >> 5;

    if (t < 32) {  // TDM-stage this chunk's 2048 code bytes (1-D tile)
        const unsigned char* src = codes + ((size_t)b * N + (size_t)c * CHUNK);
        tdm_load_tile(src, (unsigned int)(uintptr_t)&sc[0],
                      /*data_size=*/0u,
                      /*tensor_d0=*/CHUNK, /*tensor_d1=*/1u,
                      /*tile_d0=*/CHUNK, /*tile_d1=*/0u,
                      /*d0_stride=*/CHUNK);
        __builtin_amdgcn_s_wait_tensorcnt(0);
    }
    __syncthreads();

    // Phase 1: per-wave per-octant counts over its 256-point segment.
    unsigned char myc[PTS_PER_THREAD];
    unsigned int cnt[8] = {0, 0, 0, 0, 0, 0, 0, 0};
#pragma unroll
    for (int g = 0; g < PTS_PER_THREAD; ++g) {
        myc[g] = sc[w * SEG + g * 32 + lane];
#pragma unroll
        for (int o = 0; o < 8; ++o) {
            unsigned int m = __builtin_amdgcn_ballot_w32(myc[g] == (unsigned char)o);
            cnt[o] += (unsigned int)__popc(m);
        }
    }
    if (lane == 0) {
#pragma unroll
        for (int o = 0; o < 8; ++o) segCnt[w][o] = cnt[o];
    }
    __syncthreads();

    // Phase 2: segment bases = chunk offset + counts of higher segments.
    if (t < WAVES_PER_BLOCK * 8) {
        const int w2 = t >> 3;
        const int o  = t & 7;
        unsigned int base = offs[((size_t)b * 8 + o) * nChunks + c];
        for (int wp = w2 + 1; wp < WAVES_PER_BLOCK; ++wp) base += segCnt[wp][o];
        segBase[w2][o] = base;
    }
    __syncthreads();

    // Phase 3: exact descending ranks, groups high->low within the segment.
    unsigned int run[8];
#pragma unroll
    for (int o = 0; o < 8; ++o) run[o] = segBase[w][o];

    const unsigned int above = (lane == 31) ? 0u : ~((2u << lane) - 1u);
#pragma unroll
    for (int g = PTS_PER_THREAD - 1; g >= 0; --g) {
        const int oc = (int)myc[g];
        unsigned int pos = 0u;
#pragma unroll
        for (int o = 0; o < 8; ++o) {
            unsigned int m = __builtin_amdgcn_ballot_w32(oc == o);
            if (oc == o) pos = run[o] + (unsigned int)__popc(m & above);
            run[o] += (unsigned int)__popc(m);
        }
        const unsigned int gi = (unsigned int)(c * CHUNK + w * SEG + g * 32 + lane);
        out[((size_t)b * 8 + oc) * N + pos] = (long long)gi;
    }
}

// ---------------------------------------------------------------------------
extern "C" void kernel_launch(void* const* d_in, const int* in_sizes, int n_in,
                              void* d_out, int out_size, void* d_ws, size_t ws_size,
                              hipStream_t stream) {
    const float* pcs = (const float*)d_in[0];
    long long* out = (long long*)d_out;

    const int N = 65536;
    const int B = in_sizes[0] / (3 * N);      // 64
    const int nChunks = N / CHUNK;            // 32

    // Workspace carve-up: codes (B*N bytes), hist, offs, totals (uint each).
    unsigned char* codes = (unsigned char*)d_ws;                 // 4 MiB
    unsigned int* hist   = (unsigned int*)(codes + (size_t)B * N);
    unsigned int* offs   = hist + (size_t)B * nChunks * 8;       // 64 KiB each
    unsigned int* totals = offs + (size_t)B * nChunks * 8;       // 2 KiB

    // 1) Octant codes + per-chunk histograms (TDM staging of [3 x CHUNK]).
    dim3 gridA(nChunks, B);
    octant_code_hist_kernel<<<gridA, BLOCK, 0, stream>>>(pcs, codes, hist, N, nChunks);

    // 2) Reverse exclusive scan of chunk histograms + per-row totals.
    rev_scan_kernel<<<B, 32, 0, stream>>>(hist, offs, totals, nChunks);

    // 3) Zero only the tail [cnt, N) of each of the 512 octant rows.
    dim3 gridF(N / CHUNK, 8, B);
    tail_fill_kernel<<<gridF, BLOCK, 0, stream>>>(totals, out, N);

    // 4) Deterministic ballot-ranked scatter (TDM staging of code bytes).
    dim3 gridD(nChunks, B);
    scatter_ranks_kernel<<<gridD, BLOCK, 0, stream>>>(codes, offs, out, N, nChunks);
}